// TetGCN_6279242187228
// MI455X (gfx1250) — compile-verified
//
#include <hip/hip_runtime.h>
#include <math.h>

#define MAX_DELTA_LOG 0.3f
#define EPS_F 1e-8
#define HDIM 64
#define NB_RED 512

typedef __attribute__((ext_vector_type(2))) float v2f;
typedef __attribute__((ext_vector_type(8))) float v8f;

// ---------------- Kernel 1: per-block partial sum / sumsq (f64) ----------------
__global__ __launch_bounds__(256) void k_reduce(const float* __restrict__ hu, int n,
                                                double* __restrict__ psum,
                                                double* __restrict__ psq) {
  __shared__ double ssum[256];
  __shared__ double ssq[256];
  double s = 0.0, q = 0.0;
  for (int i = blockIdx.x * blockDim.x + threadIdx.x; i < n; i += gridDim.x * blockDim.x) {
    double x = (double)hu[i];
    s += x;
    q += x * x;
  }
  int t = threadIdx.x;
  ssum[t] = s; ssq[t] = q;
  __syncthreads();
  for (int off = 128; off > 0; off >>= 1) {
    if (t < off) { ssum[t] += ssum[t + off]; ssq[t] += ssq[t + off]; }
    __syncthreads();
  }
  if (t == 0) { psum[blockIdx.x] = ssum[0]; psq[blockIdx.x] = ssq[0]; }
}

// ---------------- Kernel 2: finalize mu and 1/(std+eps) ----------------
__global__ __launch_bounds__(NB_RED) void k_finalize(const double* __restrict__ psum,
                                                     const double* __restrict__ psq,
                                                     int n, float* __restrict__ stats) {
  __shared__ double ssum[NB_RED];
  __shared__ double ssq[NB_RED];
  int t = threadIdx.x;
  ssum[t] = psum[t];
  ssq[t] = psq[t];
  __syncthreads();
  for (int off = NB_RED / 2; off > 0; off >>= 1) {
    if (t < off) { ssum[t] += ssum[t + off]; ssq[t] += ssq[t + off]; }
    __syncthreads();
  }
  if (t == 0) {
    double s = ssum[0], q = ssq[0];
    double mu = s / (double)n;
    double var = (q - s * mu) / (double)(n - 1);   // unbiased (ddof=1)
    if (var < 0.0) var = 0.0;
    double sigma = sqrt(var) + EPS_F;
    stats[0] = (float)mu;
    stats[1] = (float)(1.0 / sigma);
  }
}

// ---------------- Kernel 3: per-node hidden layer via WMMA f32 16x16x4 ----------------
// s_nei[i] = relu(b1 + a_i*Wn1 + c_i*Ws1) . Wn2
// s_self[i] = relu(b1 + a_i*Wn1 + c_i*Ws1) . Ws2
// where a_i = (sum_nbr hu - 4mu)/sigma, c_i = (hu[i]-mu)/sigma
__global__ __launch_bounds__(256) void k_hidden(
    const float* __restrict__ hu, const int* __restrict__ nidx,
    const float* __restrict__ Wn1, const float* __restrict__ Ws1,
    const float* __restrict__ b1, const float* __restrict__ Wn2,
    const float* __restrict__ Ws2, const float* __restrict__ stats,
    float* __restrict__ s_nei, float* __restrict__ s_self, int n) {
  const float mu = stats[0];
  const float isg = stats[1];
  const int lane = threadIdx.x & 31;
  const int base = (blockIdx.x * (blockDim.x >> 5) + (threadIdx.x >> 5)) * 32;
  const int i = base + lane;

  // per-node features (a = normalized neighbor sum, c = normalized self)
  float a_f = 0.f, c_f = 0.f;
  if (i < n) {
    float self = hu[i];
    const int* __restrict__ e = nidx + 4 * i;
    float s0 = hu[e[0]] + hu[e[1]] + hu[e[2]] + hu[e[3]];
    a_f = (s0 - 4.f * mu) * isg;
    c_f = (self - mu) * isg;
  }

#if __has_builtin(__builtin_amdgcn_wmma_f32_16x16x4_f32)
  const int col = lane & 15;
  const bool lo = lane < 16;

  // B (4x16 per 16-wide hidden chunk): rows K0=Wn1, K1=Ws1, K2=K3=0.
  // Lanes 0-15 carry K0/K1 in vgpr0/vgpr1; upper half zeroed (K2,K3 rows are zero).
  v2f Bm[4];
  float bia[4], wn2c[4], ws2c[4];
#pragma unroll
  for (int c = 0; c < 4; ++c) {
    int k = c * 16 + col;
    Bm[c].x = lo ? Wn1[k] : 0.f;
    Bm[c].y = lo ? Ws1[k] : 0.f;
    bia[c] = b1[k];
    wn2c[c] = Wn2[k];
    ws2c[c] = Ws2[k];
  }

  // Tile0: nodes base+0..15 (features live in lanes 0-15).
  // Tile1: nodes base+16..31 (pull features down from lanes 16-31).
  float a_hi = __shfl(a_f, col + 16, 32);
  float c_hi = __shfl(c_f, col + 16, 32);
  v2f A0, A1;
  A0.x = lo ? a_f : 0.f;  A0.y = lo ? c_f : 0.f;
  A1.x = lo ? a_hi : 0.f; A1.y = lo ? c_hi : 0.f;

  float an0[8], as0[8], an1[8], as1[8];
#pragma unroll
  for (int v = 0; v < 8; ++v) { an0[v] = as0[v] = an1[v] = as1[v] = 0.f; }

#pragma unroll
  for (int c = 0; c < 4; ++c) {
    v8f cz = {0.f, 0.f, 0.f, 0.f, 0.f, 0.f, 0.f, 0.f};
    v8f d0 = __builtin_amdgcn_wmma_f32_16x16x4_f32(false, A0, false, Bm[c],
                                                   (short)0, cz, false, false);
    v8f d1 = __builtin_amdgcn_wmma_f32_16x16x4_f32(false, A1, false, Bm[c],
                                                   (short)0, cz, false, false);
#pragma unroll
    for (int v = 0; v < 8; ++v) {
      float r0 = fmaxf(d0[v] + bia[c], 0.f);
      float r1 = fmaxf(d1[v] + bia[c], 0.f);
      an0[v] = fmaf(wn2c[c], r0, an0[v]);
      as0[v] = fmaf(ws2c[c], r0, as0[v]);
      an1[v] = fmaf(wn2c[c], r1, an1[v]);
      as1[v] = fmaf(ws2c[c], r1, as1[v]);
    }
  }

  // Reduce the 16 hidden columns: xor masks 1,2,4,8 stay inside each 16-lane half.
  // Lanes 0-15 end with node row v, lanes 16-31 with node row v+8.
  const int half8 = lo ? 0 : 8;
#pragma unroll
  for (int v = 0; v < 8; ++v) {
    float rn0 = an0[v], rs0 = as0[v], rn1 = an1[v], rs1 = as1[v];
#pragma unroll
    for (int m = 8; m >= 1; m >>= 1) {
      rn0 += __shfl_xor(rn0, m, 32);
      rs0 += __shfl_xor(rs0, m, 32);
      rn1 += __shfl_xor(rn1, m, 32);
      rs1 += __shfl_xor(rs1, m, 32);
    }
    if (col == v) {
      int n0 = base + v + half8;            // tile0 node
      if (n0 < n) { s_nei[n0] = rn0; s_self[n0] = rs0; }
      int n1 = base + 16 + v + half8;       // tile1 node
      if (n1 < n) { s_nei[n1] = rn1; s_self[n1] = rs1; }
    }
  }
#else
  // Scalar fallback (should not trigger on gfx1250 toolchains)
  float sn = 0.f, ss = 0.f;
#pragma unroll 4
  for (int k = 0; k < HDIM; ++k) {
    float pre = b1[k] + a_f * Wn1[k] + c_f * Ws1[k];
    float r = fmaxf(pre, 0.f);
    sn = fmaf(Wn2[k], r, sn);
    ss = fmaf(Ws2[k], r, ss);
  }
  if (i < n) { s_nei[i] = sn; s_self[i] = ss; }
#endif
}

// ---------------- Kernel 4: scalar gather + tanh ----------------
__global__ __launch_bounds__(256) void k_output(const int* __restrict__ nidx,
                                                const float* __restrict__ s_nei,
                                                const float* __restrict__ s_self,
                                                const float* __restrict__ b2,
                                                float* __restrict__ out, int n) {
  int i = blockIdx.x * blockDim.x + threadIdx.x;
  if (i >= n) return;
  const int* __restrict__ e = nidx + 4 * i;
  float acc = b2[0] + s_self[i] + s_nei[e[0]] + s_nei[e[1]] + s_nei[e[2]] + s_nei[e[3]];
  out[i] = MAX_DELTA_LOG * tanhf(acc);
}

extern "C" void kernel_launch(void* const* d_in, const int* in_sizes, int n_in,
                              void* d_out, int out_size, void* d_ws, size_t ws_size,
                              hipStream_t stream) {
  const float* hu  = (const float*)d_in[0];
  const int*   nid = (const int*)d_in[1];
  // d_in[2] neighbor_offsets unused: fixed degree-4 CSR (arange*4)
  const float* Wn1 = (const float*)d_in[3];
  const float* Ws1 = (const float*)d_in[4];
  const float* b1  = (const float*)d_in[5];
  const float* Wn2 = (const float*)d_in[6];
  const float* Ws2 = (const float*)d_in[7];
  const float* b2  = (const float*)d_in[8];
  float* out = (float*)d_out;
  const int n = in_sizes[0];

  // Workspace layout (written fresh every call; no reliance on prior state):
  //   [0,4096)      : double psum[512]
  //   [4096,8192)   : double psq[512]
  //   [8192,8448)   : float stats[2] (mu, 1/(std+eps))
  //   [8448, +4n)   : float s_nei[n]
  //   [.., +4n)     : float s_self[n]
  char* ws = (char*)d_ws;
  double* psum = (double*)(ws);
  double* psq  = (double*)(ws + 4096);
  float* stats = (float*)(ws + 8192);
  float* s_nei = (float*)(ws + 8448);
  size_t n_pad = ((size_t)n + 63) & ~(size_t)63;
  float* s_self = (float*)(ws + 8448 + 4 * n_pad);

  k_reduce<<<NB_RED, 256, 0, stream>>>(hu, n, psum, psq);
  k_finalize<<<1, NB_RED, 0, stream>>>(psum, psq, n, stats);
  int nb = (n + 255) / 256;
  k_hidden<<<nb, 256, 0, stream>>>(hu, nid, Wn1, Ws1, b1, Wn2, Ws2, stats, s_nei, s_self, n);
  k_output<<<nb, 256, 0, stream>>>(nid, s_nei, s_self, b2, out, n);
}